// SeparableConvFlowModule_47253230191039
// MI455X (gfx1250) — compile-verified
//
#include <hip/hip_runtime.h>
#include <hip/hip_bf16.h>
#include <stdint.h>

// ---- problem constants (match reference) ----
constexpr int Hc = 384, Wc = 384, HW = Hc * Wc;   // 147456 pixels/plane
constexpr int Bc = 4;
constexpr int F  = 51;                            // taps
constexpr float CENTER = 25.0f;                   // (F-1)/2

// ---- tiling ----
constexpr int TILE_PIX        = 32;               // pixels per wave-tile
constexpr int TILES_PER_PLANE = HW / TILE_PIX;    // 4608
constexpr int WAVES_PER_BLOCK = 4;                // 128 threads/block
constexpr int ROWSTRIDE       = 32;               // floats per tile row (128 B)
constexpr int TROWS           = 52;               // 51 taps + 1 zero pad row
constexpr int TILE_FLOATS     = TROWS * ROWSTRIDE;        // 1664 floats = 6656 B
constexpr int LDS_FLOATS      = WAVES_PER_BLOCK * 2 * TILE_FLOATS; // 53248 B
constexpr int KSTEPS          = 13;               // ceil(51/4), K padded to 52

#if defined(__gfx1250__) && __has_builtin(__builtin_amdgcn_global_load_async_to_lds_b128)
#define ASYNC_LDS 1
#else
#define ASYNC_LDS 0
#endif

typedef __attribute__((ext_vector_type(2))) float v2f;
typedef __attribute__((ext_vector_type(8))) float v8f;

#if ASYNC_LDS
typedef int v4i __attribute__((ext_vector_type(4)));
typedef __attribute__((address_space(1))) v4i as1_v4i;   // global
typedef __attribute__((address_space(3))) v4i as3_v4i;   // LDS
#endif

// Stage one 51-row x 32-pixel f32 tile (rows stride HW in global) into LDS.
// 51 rows * 128B = 408 b128 lane-chunks -> 13 wave-wide async issues.
__device__ __forceinline__ void stage_tile(const float* __restrict__ g,
                                           float* __restrict__ tile, int lane) {
#pragma unroll
  for (int it = 0; it < 13; ++it) {
    int chunk = it * 32 + lane;
    if (chunk < 408) {
      int row = chunk >> 3;          // tap row 0..50
      int seg = chunk & 7;           // 16B segment within the 128B row
      const float* gp = g + (size_t)row * HW + seg * 4;
      float*       lp = tile + row * ROWSTRIDE + seg * 4;
#if ASYNC_LDS
      __builtin_amdgcn_global_load_async_to_lds_b128(
          (as1_v4i*)gp, (as3_v4i*)lp, /*offset=*/0, /*cpol=*/0);
#else
      *(float4*)lp = *(const float4*)gp;   // synchronous fallback
#endif
    }
  }
}

__device__ __forceinline__ void wait_async() {
#if ASYNC_LDS
#if __has_builtin(__builtin_amdgcn_s_wait_asynccnt)
  __builtin_amdgcn_s_wait_asynccnt(0);
#else
  asm volatile("s_wait_asynccnt 0" ::: "memory");
#endif
#endif
  asm volatile("" ::: "memory");   // keep LDS reads below the drain
}

// D(16x16) = A(16x4) x B(4x16) + C chained over K = 0..51 (52, last col zeroed).
// A row0 = ones, row1 = tap index, rows 2..15 = 0. afrag[] is precomputed once
// per wave and shared by all four reduction chains.
// Lanes 0..15 of the result hold D[0][n]=sum in c[0], D[1][n]=wsum in c[1].
__device__ __forceinline__ v8f reduce_tile(const float* __restrict__ tile,
                                           const v2f* __restrict__ afrag,
                                           int lane, int ntile) {
  const int h   = lane >> 4;        // lane-half selects K pair (0,1) vs (2,3)
  const int col = ntile * 16 + (lane & 15);
  v8f c = {};
#pragma unroll
  for (int i = 0; i < KSTEPS; ++i) {
    const int kg0 = i * 4 + 2 * h;
    v2f b;
    b.x = tile[kg0 * ROWSTRIDE + col];         // row 51 was zeroed in LDS
    b.y = tile[(kg0 + 1) * ROWSTRIDE + col];
    c = __builtin_amdgcn_wmma_f32_16x16x4_f32(
        /*neg_a=*/false, afrag[i], /*neg_b=*/false, b,
        /*c_mod=*/(short)0, c, /*reuse_a=*/false, /*reuse_b=*/false);
  }
  return c;
}

__global__ __launch_bounds__(WAVES_PER_BLOCK * 32)
void SeparableConvFlow_kernel(const float* __restrict__ in2,   // [B,51,H,W] vertical taps
                              const float* __restrict__ in3,   // [B,51,H,W] horizontal taps
                              float* __restrict__ out) {       // [B,2,H,W]
  __shared__ __align__(16) float lds[LDS_FLOATS];

  const int lane = threadIdx.x & 31;
  const int widx = threadIdx.x >> 5;
  const int wave = blockIdx.x * WAVES_PER_BLOCK + widx;
  const int b    = wave / TILES_PER_PLANE;
  const int t    = wave % TILES_PER_PLANE;
  const int pix  = t * TILE_PIX;

  float* tile3 = &lds[widx * 2 * TILE_FLOATS];
  float* tile2 = tile3 + TILE_FLOATS;
  const float* g3 = in3 + (size_t)b * F * HW + pix;
  const float* g2 = in2 + (size_t)b * F * HW + pix;

  // Issue both tensors' DMA before waiting -> overlapped fills.
  stage_tile(g3, tile3, lane);
  stage_tile(g2, tile2, lane);
  // Zero the K-pad row (prevents 0*NaN poisoning the WMMA accumulator).
  tile3[51 * ROWSTRIDE + lane] = 0.0f;
  tile2[51 * ROWSTRIDE + lane] = 0.0f;

  // Precompute the 13 A-fragments ONCE (shared by all 4 WMMA chains):
  //   A[m][k] = (m==0) ? 1 : (m==1) ? k : 0, with k==51 column zeroed.
  // Per lane: a = sel1 + selw*k  (one FMA per component).
  const int   m    = lane & 15;
  const int   h    = lane >> 4;
  const float sel1 = (m == 0) ? 1.0f : 0.0f;
  const float selw = (m == 1) ? 1.0f : 0.0f;
  v2f afrag[KSTEPS];
#pragma unroll
  for (int i = 0; i < KSTEPS; ++i) {
    const int kg0 = i * 4 + 2 * h;
    const int kg1 = kg0 + 1;
    afrag[i].x = (kg0 < F) ? fmaf(selw, (float)kg0, sel1) : 0.0f;
    afrag[i].y = (kg1 < F) ? fmaf(selw, (float)kg1, sel1) : 0.0f;
  }

  wait_async();

  // flow_x from input3, flow_y from input2 (two 16-pixel N-tiles each).
  v8f cx0 = reduce_tile(tile3, afrag, lane, 0);
  v8f cx1 = reduce_tile(tile3, afrag, lane, 1);
  v8f cy0 = reduce_tile(tile2, afrag, lane, 0);
  v8f cy1 = reduce_tile(tile2, afrag, lane, 1);

  if (lane < 16) {
    float* ob = out + (size_t)b * 2 * HW;
    ob[pix + lane]           = cx0[1] / cx0[0] - CENTER;
    ob[pix + 16 + lane]      = cx1[1] / cx1[0] - CENTER;
    ob[HW + pix + lane]      = cy0[1] / cy0[0] - CENTER;
    ob[HW + pix + 16 + lane] = cy1[1] / cy1[0] - CENTER;
  }
}

extern "C" void kernel_launch(void* const* d_in, const int* in_sizes, int n_in,
                              void* d_out, int out_size, void* d_ws, size_t ws_size,
                              hipStream_t stream) {
  (void)in_sizes; (void)n_in; (void)out_size; (void)d_ws; (void)ws_size;
  // setup_inputs order: input1 (unused), input2 (vertical), input3 (horizontal)
  const float* in2 = (const float*)d_in[1];
  const float* in3 = (const float*)d_in[2];
  float* out = (float*)d_out;

  const int total_waves = Bc * TILES_PER_PLANE;            // 18432
  const int blocks      = total_waves / WAVES_PER_BLOCK;   // 4608
  SeparableConvFlow_kernel<<<blocks, WAVES_PER_BLOCK * 32, 0, stream>>>(in2, in3, out);
}